// GAT_24558622998850
// MI455X (gfx1250) — compile-verified
//
#include <hip/hip_runtime.h>
#include <hip/hip_bf16.h>

typedef __attribute__((ext_vector_type(2))) float v2f;
typedef __attribute__((ext_vector_type(8))) float v8f;

#define LEAKY 0.2f

// ---------------------------------------------------------------------------
// Order-preserving encoding of float into uint so segment-max can use
// hardware global_atomic_max_u32 (all real floats encode > 0, so zero-init ok)
// ---------------------------------------------------------------------------
__device__ __forceinline__ unsigned enc_f32(float f) {
  unsigned u = __float_as_uint(f);
  return (u >> 31) ? ~u : (u | 0x80000000u);
}
__device__ __forceinline__ float dec_f32(unsigned k) {
  return (k & 0x80000000u) ? __uint_as_float(k & 0x7FFFFFFFu)
                           : __uint_as_float(~k);
}

__device__ __forceinline__ void atomic_add_f32(float* p, float v) {
  __hip_atomic_fetch_add(p, v, __ATOMIC_RELAXED, __HIP_MEMORY_SCOPE_AGENT);
}

// ---------------------------------------------------------------------------
// Zero fill
// ---------------------------------------------------------------------------
__global__ void k_fill_zero(float* __restrict__ p, int n) {
  int i = blockIdx.x * blockDim.x + threadIdx.x;
  if (i < n) p[i] = 0.0f;
}

// ---------------------------------------------------------------------------
// Pack W1 (H,F,D) directly into WMMA B-fragment order.
// Fragment layout: Bp[((ct*32 + ks)*32 + lane)*2 + j] where
//   f   = ks*4 + (lane>>4)*2 + j        (K index)
//   col = ct*16 + (lane&15)             (N index); h = col>>5, d = col&31
// Per k-step each wave then reads one contiguous 256B chunk (b64 per lane).
// ---------------------------------------------------------------------------
__global__ void k_pack_w1_frag(const float* __restrict__ w,
                               float* __restrict__ bp, int H, int F, int D,
                               int M) {
  int o = blockIdx.x * blockDim.x + threadIdx.x;
  if (o >= F * M) return;
  int j    = o & 1;
  int lane = (o >> 1) & 31;
  int ks   = (o >> 6) & 31;       // F/4 = 32 k-steps
  int ct   = o >> 11;             // 64 floats per k-step, 2048 per col tile
  int f    = ks * 4 + (lane >> 4) * 2 + j;
  int col  = ct * 16 + (lane & 15);
  int h    = col / D;
  int d    = col - h * D;
  bp[o] = w[(h * F + f) * D + d];
}

// Generic fragment pack of a row-major B[K x M] (used for W2: 128 x 32)
__global__ void k_pack_b_frag(const float* __restrict__ b,
                              float* __restrict__ bp, int K, int M) {
  int o = blockIdx.x * blockDim.x + threadIdx.x;
  if (o >= K * M) return;
  int j    = o & 1;
  int lane = (o >> 1) & 31;
  int ks   = (o >> 6) % (K / 4);
  int ct   = o / (64 * (K / 4));
  int f    = ks * 4 + (lane >> 4) * 2 + j;
  int col  = ct * 16 + (lane & 15);
  bp[o] = b[f * M + col];
}

// ---------------------------------------------------------------------------
// WMMA GEMM: C[N x M] = A[N x 128] * B[128 x M], f32, V_WMMA_F32_16X16X4_F32.
// One block computes a 16-row stripe; wave w owns the 16-col tile at w*16.
// blockDim.x = 32 * (M/16).  A tile staged in LDS (stride K+1 -> bank =
// (row + col) mod 64 varies across lanes).  B pre-packed in fragment order:
// one coalesced global_load_b64 per lane per WMMA step.
// ---------------------------------------------------------------------------
__global__ void k_gemm_wmma_f32(const float* __restrict__ A,
                                const float* __restrict__ Bp,
                                float* __restrict__ C, int M) {
  constexpr int K = 128;
  constexpr int LDA = K + 1;
  __shared__ float As[16 * LDA];

  const int tid  = threadIdx.x;
  const int row0 = blockIdx.x * 16;

  // cooperative load of the shared 16x128 A tile
  for (int i = tid; i < 16 * K; i += blockDim.x) {
    int r = i >> 7;
    int c = i & 127;
    As[r * LDA + c] = A[(row0 + r) * K + c];
  }
  __syncthreads();

  const int wave = tid >> 5;
  const int lane = tid & 31;
  const int half = lane >> 4;   // 0: K pair {0,1}/rows 0..7 ; 1: {2,3}/rows 8..15
  const int l15  = lane & 15;
  const int koff = half * 2;
  const int col0 = wave * 16;

  // per-wave fragment stream: 64 floats per k-step, 32 k-steps per col tile
  const float* bptr = Bp + (size_t)wave * (K / 4) * 64 + lane * 2;

  v8f acc = {};
#pragma unroll
  for (int ks = 0; ks < K / 4; ++ks) {
    const int k = ks * 4;
    v2f a, b;
    a.x = As[l15 * LDA + k + koff];
    a.y = As[l15 * LDA + k + koff + 1];
    b = *(const v2f*)(bptr + ks * 64);
    acc = __builtin_amdgcn_wmma_f32_16x16x4_f32(
        /*neg_a=*/false, a, /*neg_b=*/false, b,
        /*c_mod=*/(short)0, acc, /*reuse_a=*/false, /*reuse_b=*/false);
  }

  const int crow0 = row0 + half * 8;
  const int ccol  = col0 + l15;
#pragma unroll
  for (int i = 0; i < 8; ++i)
    C[(crow0 + i) * M + ccol] = acc[i];
}

// ---------------------------------------------------------------------------
// es[n,h] = z[n,h,:] . a[h, 0:D] ;  ed[n,h] = z[n,h,:] . a[h, D:2D]
// ---------------------------------------------------------------------------
__global__ void k_attn_coef(const float* __restrict__ z,
                            const float* __restrict__ a,
                            float* __restrict__ es, float* __restrict__ ed,
                            int N, int H, int D) {
  int gid = blockIdx.x * blockDim.x + threadIdx.x;
  if (gid >= N * H) return;
  int n = gid / H;
  int h = gid - n * H;
  const float* zp = z + ((long long)n * H + h) * D;
  const float* as = a + h * 2 * D;
  const float* ad = as + D;
  float s = 0.f, d = 0.f;
  for (int i = 0; i < D; ++i) {
    float zv = zp[i];
    s += zv * as[i];
    d += zv * ad[i];
  }
  es[gid] = s;
  ed[gid] = d;
}

// ---------------------------------------------------------------------------
// Edge pass 1: e = leaky_relu(es[src]+ed[dst]); segment-max via atomic max
// ---------------------------------------------------------------------------
__global__ void k_edge_logits(const float* __restrict__ es,
                              const float* __restrict__ ed,
                              const int* __restrict__ src,
                              const int* __restrict__ dst,
                              float* __restrict__ elog,
                              unsigned* __restrict__ menc, int E, int H) {
  int e = blockIdx.x * blockDim.x + threadIdx.x;
  if (e >= E) return;
  int s = src[e], t = dst[e];
  for (int h = 0; h < H; ++h) {
    float v = es[s * H + h] + ed[t * H + h];
    v = v > 0.f ? v : LEAKY * v;
    elog[(long long)e * H + h] = v;
    atomicMax(&menc[t * H + h], enc_f32(v));
  }
}

// ---------------------------------------------------------------------------
// Edge pass 2: ex = exp(e - m[dst]); segment-sum denominators
// ---------------------------------------------------------------------------
__global__ void k_edge_exp(const int* __restrict__ dst,
                           const unsigned* __restrict__ menc,
                           float* __restrict__ elog,
                           float* __restrict__ denom, int E, int H) {
  int e = blockIdx.x * blockDim.x + threadIdx.x;
  if (e >= E) return;
  int t = dst[e];
  for (int h = 0; h < H; ++h) {
    float m = dec_f32(menc[t * H + h]);
    float x = expf(elog[(long long)e * H + h] - m);
    elog[(long long)e * H + h] = x;
    atomic_add_f32(&denom[t * H + h], x);
  }
}

// ---------------------------------------------------------------------------
// Edge pass 3: out[dst] += (ex/denom[dst]) * z[src]; one thread per
// (edge, feature) so z gathers and atomic adds are coalesced per edge row.
// ---------------------------------------------------------------------------
__global__ void k_edge_aggregate(const int* __restrict__ src,
                                 const int* __restrict__ dst,
                                 const float* __restrict__ ex,
                                 const float* __restrict__ den,
                                 const float* __restrict__ z,
                                 float* __restrict__ out, int E, int H, int D) {
  int HD = H * D;
  long long gid = (long long)blockIdx.x * blockDim.x + threadIdx.x;
  long long total = (long long)E * HD;
  if (gid >= total) return;
  int e = (int)(gid / HD);
  int c = (int)(gid - (long long)e * HD);
  int h = c / D;
  int s = src[e], t = dst[e];
  float alpha = ex[(long long)e * H + h] / den[t * H + h];
  float v = alpha * z[(long long)s * HD + c];
  atomic_add_f32(&out[(long long)t * HD + c], v);
}

// ---------------------------------------------------------------------------
// Launch
// ---------------------------------------------------------------------------
static inline int cdiv(long long a, int b) { return (int)((a + b - 1) / b); }

extern "C" void kernel_launch(void* const* d_in, const int* in_sizes, int n_in,
                              void* d_out, int out_size, void* d_ws, size_t ws_size,
                              hipStream_t stream) {
  const float* feat = (const float*)d_in[0];
  const float* W1   = (const float*)d_in[1];
  const float* a1   = (const float*)d_in[2];
  const float* W2   = (const float*)d_in[3];
  const float* a2   = (const float*)d_in[4];
  const int*   src  = (const int*)d_in[5];
  const int*   dst  = (const int*)d_in[6];
  float*       out  = (float*)d_out;

  const int F = 128, H1 = 4, D1 = 32, D2 = 32;
  const int HD1 = H1 * D1;                 // 128
  const int N = in_sizes[0] / F;           // 50000
  const int E = in_sizes[5];               // 800000

  // ---- workspace carve ----
  float* ws    = (float*)d_ws;
  float* bpk1  = ws;                 ws += F * HD1;            // 128x128 frag
  float* bpk2  = ws;                 ws += F * D2;             // 128x32 frag
  float* z1    = ws;                 ws += (long long)N * HD1; // N x 128
  float* es1   = ws;                 ws += N * H1;
  float* ed1   = ws;                 ws += N * H1;
  float* elog1 = ws;                 ws += (long long)E * H1;  // also ex1
  unsigned* m1 = (unsigned*)ws;      ws += N * H1;
  float* den1  = ws;                 ws += N * H1;
  float* h1    = ws;                 ws += (long long)N * HD1; // N x 128
  float* z2    = ws;                 ws += (long long)N * D2;  // N x 32
  float* es2   = ws;                 ws += N;
  float* ed2   = ws;                 ws += N;
  float* elog2 = ws;                 ws += E;
  unsigned* m2 = (unsigned*)ws;      ws += N;
  float* den2  = ws;                 ws += N;

  const int TB = 256;

  // ---- zero accumulation targets (enc(-inf) > 0, so zero is a valid -inf) ----
  k_fill_zero<<<cdiv((long long)N * HD1, TB), TB, 0, stream>>>(h1, N * HD1);
  k_fill_zero<<<cdiv(N * H1, TB), TB, 0, stream>>>((float*)m1, N * H1);
  k_fill_zero<<<cdiv(N * H1, TB), TB, 0, stream>>>(den1, N * H1);
  k_fill_zero<<<cdiv(N, TB), TB, 0, stream>>>((float*)m2, N);
  k_fill_zero<<<cdiv(N, TB), TB, 0, stream>>>(den2, N);
  k_fill_zero<<<cdiv((long long)N * D2, TB), TB, 0, stream>>>(out, N * D2);

  // ---- weight packing into WMMA fragment order ----
  k_pack_w1_frag<<<cdiv(F * HD1, TB), TB, 0, stream>>>(W1, bpk1, H1, F, D1, HD1);
  k_pack_b_frag<<<cdiv(F * D2, TB), TB, 0, stream>>>(W2, bpk2, F, D2);

  // ---- layer 1 ----
  k_gemm_wmma_f32<<<N / 16, 32 * (HD1 / 16), 0, stream>>>(feat, bpk1, z1, HD1);
  k_attn_coef<<<cdiv(N * H1, TB), TB, 0, stream>>>(z1, a1, es1, ed1, N, H1, D1);
  k_edge_logits<<<cdiv(E, TB), TB, 0, stream>>>(es1, ed1, src, dst, elog1, m1, E, H1);
  k_edge_exp<<<cdiv(E, TB), TB, 0, stream>>>(dst, m1, elog1, den1, E, H1);
  k_edge_aggregate<<<cdiv((long long)E * HD1, TB), TB, 0, stream>>>(
      src, dst, elog1, den1, z1, h1, E, H1, D1);

  // ---- layer 2 (H=1, D=32) ----
  k_gemm_wmma_f32<<<N / 16, 32 * (D2 / 16), 0, stream>>>(h1, bpk2, z2, D2);
  k_attn_coef<<<cdiv(N, TB), TB, 0, stream>>>(z2, a2, es2, ed2, N, 1, D2);
  k_edge_logits<<<cdiv(E, TB), TB, 0, stream>>>(es2, ed2, src, dst, elog2, m2, E, 1);
  k_edge_exp<<<cdiv(E, TB), TB, 0, stream>>>(dst, m2, elog2, den2, E, 1);
  k_edge_aggregate<<<cdiv((long long)E * D2, TB), TB, 0, stream>>>(
      src, dst, elog2, den2, z2, out, E, 1, D2);
}